// ROIPooling_74869869904205
// MI455X (gfx1250) — compile-verified
//
#include <hip/hip_runtime.h>
#include <hip/hip_bf16.h>
#include <stdint.h>

// ROI align (crop_and_resize to 14x14) + 2x2 maxpool -> (1024,7,7,256) fp32.
//
// Roofline: feature maps (5.1 MB) are L2-resident (192 MB); ~800 MB of
// coalesced b128 gathers from L2, ~57 MB HBM traffic -> the kernel is
// L2-gather/latency bound. No shared-B contraction exists (each sample's
// 4-corner "B matrix" is unique per output row), so WMMA cannot apply
// without inflating FLOPs ~12x at the same gather cost.
//
// CDNA5 path used instead: async global->LDS engine (ASYNCcnt) with a
// triple-buffered, distance-2 software pipeline across pooled cells —
// async loads carry no VGPR destinations and complete in order, so
// s_wait_asynccnt <= N retires exactly one cell's 16-transfer batch.

namespace {
constexpr int kH = 50, kW = 50, kC = 256, kR = 512;
constexpr int kPH = 7, kPW = 7, kK = 2;
constexpr int kCH = kPH * kK;   // 14: crop grid size
constexpr int kQ  = kC / 4;     // 64 float4 channel-quads = blockDim (2 wave32s)
}

__device__ __forceinline__ float4 f4_lerp(const float4 a, const float4 b, const float w) {
  const float u = 1.0f - w;
  return make_float4(a.x * u + b.x * w, a.y * u + b.y * w,
                     a.z * u + b.z * w, a.w * u + b.w * w);
}
__device__ __forceinline__ float4 f4_max(const float4 a, const float4 b) {
  return make_float4(fmaxf(a.x, b.x), fmaxf(a.y, b.y),
                     fmaxf(a.z, b.z), fmaxf(a.w, b.w));
}

__global__ __launch_bounds__(kQ) void roi_pool_async_kernel(
    const float* __restrict__ fm, const float* __restrict__ rois,
    float* __restrict__ out, const int nroi)
{
  const int py = blockIdx.x % kPH;
  const int n  = blockIdx.x / kPH;
  if (n >= nroi) return;
  const int lane = threadIdx.x;          // channel quad 0..63
  const int b    = n / kR;

  const float ry1 = rois[(size_t)n * 4 + 0];
  const float rx1 = rois[(size_t)n * 4 + 1];
  const float ry2 = rois[(size_t)n * 4 + 2];
  const float rx2 = rois[(size_t)n * 4 + 3];
  const float ybase = ry1 * (float)(kH - 1);
  const float xbase = rx1 * (float)(kW - 1);
  const float ystep = (ry2 - ry1) * (float)(kH - 1) / (float)(kCH - 1);
  const float xstep = (rx2 - rx1) * (float)(kW - 1) / (float)(kCH - 1);

  const float* fmb = fm + (size_t)b * (kH * kW * kC);

  // y-side state is invariant across the 7 cells of this pooled row.
  int   yrow[2][2];   // [ky][corner] -> clamped row * W
  float wy[2];
  bool  vy[2];
#pragma unroll
  for (int ky = 0; ky < 2; ++ky) {
    const float iny = ybase + (float)(kK * py + ky) * ystep;
    const float y0f = floorf(iny);
    const int   y0  = (int)y0f;
    yrow[ky][0] = min(max(y0,     0), kH - 1) * kW;
    yrow[ky][1] = min(max(y0 + 1, 0), kH - 1) * kW;
    wy[ky] = iny - y0f;
    vy[ky] = (iny >= 0.0f) && (iny <= (float)(kH - 1));
  }

  // 3 buffers x 16 corner rows x 64 quads x 16B = 48 KB LDS (320 KB WGP).
  __shared__ float4 stage[3][16][kQ];

  // Issue the 16 async corner gathers for pooled cell px into buffer `buf`.
  auto issue_cell = [&](int buf, int px) {
    int xcol[2][2];   // [kx][corner] -> clamped column
#pragma unroll
    for (int kx = 0; kx < 2; ++kx) {
      const float inx = xbase + (float)(kK * px + kx) * xstep;
      const int   x0  = (int)floorf(inx);
      xcol[kx][0] = min(max(x0,     0), kW - 1);
      xcol[kx][1] = min(max(x0 + 1, 0), kW - 1);
    }
#pragma unroll
    for (int s = 0; s < 4; ++s) {                 // 2x2 pooled samples
      const int ky = s >> 1, kx = s & 1;
#pragma unroll
      for (int c = 0; c < 4; ++c) {               // f00,f01,f10,f11
        const float4* gsrc = reinterpret_cast<const float4*>(
            fmb + (size_t)(yrow[ky][c >> 1] + xcol[kx][c & 1]) * kC) + lane;
        // addr[31:0] of the flat LDS-aperture address == LDS byte offset.
        const unsigned ldst = (unsigned)(uintptr_t)(&stage[buf][s * 4 + c][lane]);
        asm volatile("global_load_async_to_lds_b128 %0, %1, off"
                     :: "v"(ldst), "v"(gsrc) : "memory");
      }
    }
  };

  issue_cell(0, 0);                                // pipeline prologue
  issue_cell(1, 1);

  for (int px = 0; px < kPW; ++px) {
    if (px + 2 < kPW) {
      issue_cell((px + 2) % 3, px + 2);            // prefetch distance 2
      // In-order async completion: <=32 outstanding => cell px's 16 landed.
      asm volatile("s_wait_asynccnt 0x20" ::: "memory");
    } else if (px + 1 < kPW) {
      asm volatile("s_wait_asynccnt 0x10" ::: "memory");
    } else {
      asm volatile("s_wait_asynccnt 0x0" ::: "memory");
    }
    const int buf = px % 3;

    float wx[2];
    bool  vx[2];
#pragma unroll
    for (int kx = 0; kx < 2; ++kx) {
      const float inx = xbase + (float)(kK * px + kx) * xstep;
      wx[kx] = inx - floorf(inx);
      vx[kx] = (inx >= 0.0f) && (inx <= (float)(kW - 1));
    }

    float4 best = make_float4(0.f, 0.f, 0.f, 0.f);
#pragma unroll
    for (int s = 0; s < 4; ++s) {
      const int ky = s >> 1, kx = s & 1;
      const float4 f00 = stage[buf][s * 4 + 0][lane];
      const float4 f01 = stage[buf][s * 4 + 1][lane];
      const float4 f10 = stage[buf][s * 4 + 2][lane];
      const float4 f11 = stage[buf][s * 4 + 3][lane];
      float4 v = f4_lerp(f4_lerp(f00, f01, wx[kx]),
                         f4_lerp(f10, f11, wx[kx]), wy[ky]);
      if (!(vy[ky] && vx[kx])) v = make_float4(0.f, 0.f, 0.f, 0.f);
      best = (s == 0) ? v : f4_max(best, v);       // max of the 4 samples
    }
    float4* dst = reinterpret_cast<float4*>(
        out + (((size_t)n * kPH + py) * kPW + px) * kC) + lane;
    *dst = best;                                   // coalesced b128 store
  }
}

extern "C" void kernel_launch(void* const* d_in, const int* in_sizes, int n_in,
                              void* d_out, int out_size, void* d_ws, size_t ws_size,
                              hipStream_t stream) {
  (void)n_in; (void)out_size; (void)d_ws; (void)ws_size;
  const float* fm   = (const float*)d_in[0];   // (2,50,50,256) fp32
  const float* rois = (const float*)d_in[1];   // (2,512,4)    fp32
  float* out = (float*)d_out;                  // (1024,7,7,256) fp32
  const int nroi = in_sizes[1] / 4;            // 1024
  dim3 grid(nroi * kPH);                       // one block per (roi, pooled row)
  dim3 block(kQ);                              // 64 threads = 2 wave32s
  hipLaunchKernelGGL(roi_pool_async_kernel, grid, block, 0, stream,
                     fm, rois, out, nroi);
}